// LSTM_53515292508390
// MI455X (gfx1250) — compile-verified
//
#include <hip/hip_runtime.h>
#include <math.h>

// Problem constants (match reference; n/steps taken from in_sizes at launch).
#define EMB    64
#define HID    128
#define FFD    1152
#define VSCALE 4.0f
#define XPAD   2   // keep row stride even (v2f alignment) and kill bank conflicts

typedef __attribute__((ext_vector_type(2))) float v2f;
typedef __attribute__((ext_vector_type(8))) float v8f;

// D = A(16x4,f32) * B(4x16,f32) + C(16x16,f32)  -- fp32 matrix core path
__device__ __forceinline__ v8f wmma4(v2f a, v2f b, v8f c) {
    return __builtin_amdgcn_wmma_f32_16x16x4_f32(
        /*neg_a=*/false, a, /*neg_b=*/false, b,
        /*c_mod=*/(short)0, c, /*reuse_a=*/false, /*reuse_b=*/false);
}

__device__ __forceinline__ float sigm(float x) {
    return 1.0f / (1.0f + __expf(-x));
}

// One wave (32 lanes) per 16-track tile; full 8-step scan fused in-kernel.
// All recurrent state (h, c) lives in LDS so VGPR pressure stays low
// (no spills); only the (T-1, n, 1152) output touches HBM.
__global__ __launch_bounds__(32) void lstm_encode_kernel(
    const float* __restrict__ observed,   // (T, n, 2)
    const float* __restrict__ emb_w,      // (64, 2)
    const float* __restrict__ emb_b,      // (64,)
    const float* __restrict__ w_ih,       // (512, 64)
    const float* __restrict__ w_hh,       // (512, 128)
    const float* __restrict__ b_ih,       // (512,)
    const float* __restrict__ b_hh,       // (512,)
    const float* __restrict__ ff_w,       // (1152, 128)
    const float* __restrict__ ff_b,       // (1152,)
    float* __restrict__ out,              // (T-1, n, 1152)
    int ntracks, int nsteps)
{
    __shared__ __align__(16) float xs[16][EMB + XPAD];        // x tile (A operand)
    __shared__ __align__(16) float hsbuf[2][16][HID + XPAD];  // masked h, double-buffered
    __shared__ __align__(16) float hn[16][HID + XPAD];        // unmasked h_new (ff A)
    __shared__ __align__(16) float cs[16][HID + XPAD];        // cell state
    __shared__ __align__(16) float bsum[4 * HID];             // b_ih + b_hh
    __shared__ __align__(16) float velx[16];
    __shared__ __align__(16) float vely[16];
    __shared__ unsigned char    mks[16];

    const int lane   = threadIdx.x;       // wave32
    const int m_lane = lane & 15;         // row (A/M) or col (B/N) selector
    const int lh     = lane >> 4;         // lane-half: selects K pair / M+8
    const int koff   = 2 * lh;            // K offset within a fragment
    const int track0 = blockIdx.x * 16;
    const float qnan = __uint_as_float(0x7FC00000u);

    // ---- one-time init: zero state, pre-sum biases ----
    for (int i = lane; i < 16 * (HID + XPAD); i += 32) {
        ((float*)hsbuf[0])[i] = 0.0f;
        ((float*)cs)[i]       = 0.0f;
    }
    for (int i = lane; i < 4 * HID; i += 32) bsum[i] = b_ih[i] + b_hh[i];
    __syncthreads();

    int par = 0;   // which hsbuf holds the current (previous-step) masked h

    for (int t = 0; t < nsteps; ++t) {
        // ---- 1. velocity + NaN mask (lanes 0..15, one track each) ----
        if (lane < 16) {
            int trk = track0 + lane;
            if (trk >= ntracks) trk = ntracks - 1;     // clamp; stores predicated later
            const float* p1 = observed + ((size_t)t       * ntracks + trk) * 2;
            const float* p2 = observed + ((size_t)(t + 1) * ntracks + trk) * 2;
            float a0 = p1[0], a1 = p1[1];
            float b0 = p2[0], b1 = p2[1];
            bool mk = !((a0 != a0) || (b0 != b0));     // !(isnan|isnan)
            velx[lane] = mk ? (b0 - a0) * VSCALE : 0.0f;
            vely[lane] = mk ? (b1 - a1) * VSCALE : 0.0f;
            mks[lane]  = mk ? 1 : 0;
        }
        __syncthreads();

        // ---- 2. x = vel @ emb_w.T + emb_b  (K=2: VALU, not worth a matrix op) ----
        {
            float vx = velx[m_lane], vy = vely[m_lane];
            for (int e = lh * (EMB / 2); e < (lh + 1) * (EMB / 2); ++e)
                xs[m_lane][e] = vx * emb_w[e * 2 + 0] + vy * emb_w[e * 2 + 1] + emb_b[e];
        }
        __syncthreads();

        // ---- 3. gates = x@w_ih.T + h@w_hh.T + b ; LSTM pointwise ----
        // slice s covers hidden cols n=16s..16s+15 for all four gates at once
#pragma clang loop unroll(disable)
        for (int s = 0; s < 8; ++s) {
            const int n_i = s * 16 + m_lane;
            const int n_f = n_i + HID;
            const int n_g = n_i + 2 * HID;
            const int n_o = n_i + 3 * HID;

            v8f ai, af, ag, ao;
            {
                float bi = bsum[n_i], bf = bsum[n_f], bg = bsum[n_g], bo = bsum[n_o];
#pragma unroll
                for (int r = 0; r < 8; ++r) { ai[r] = bi; af[r] = bf; ag[r] = bg; ao[r] = bo; }
            }

            // input contribution: K = EMB, one A frag shared by 4 gate WMMAs
            {
                const float* wi_p = w_ih + (size_t)n_i * EMB + koff;
                const float* wf_p = w_ih + (size_t)n_f * EMB + koff;
                const float* wg_p = w_ih + (size_t)n_g * EMB + koff;
                const float* wo_p = w_ih + (size_t)n_o * EMB + koff;
                const float* xa_p = &xs[m_lane][koff];
#pragma unroll 2
                for (int k0 = 0; k0 < EMB; k0 += 4) {
                    v2f a  = *(const v2f*)(xa_p + k0);
                    v2f wi = *(const v2f*)(wi_p + k0);
                    v2f wf = *(const v2f*)(wf_p + k0);
                    v2f wg = *(const v2f*)(wg_p + k0);
                    v2f wo = *(const v2f*)(wo_p + k0);
                    ai = wmma4(a, wi, ai);
                    af = wmma4(a, wf, af);
                    ag = wmma4(a, wg, ag);
                    ao = wmma4(a, wo, ao);
                }
            }
            // recurrent contribution: K = HID, A frags from masked h in LDS
            {
                const float* wi_p = w_hh + (size_t)n_i * HID + koff;
                const float* wf_p = w_hh + (size_t)n_f * HID + koff;
                const float* wg_p = w_hh + (size_t)n_g * HID + koff;
                const float* wo_p = w_hh + (size_t)n_o * HID + koff;
                const float* ha_p = &hsbuf[par][m_lane][koff];
#pragma unroll 2
                for (int k0 = 0; k0 < HID; k0 += 4) {
                    v2f a  = *(const v2f*)(ha_p + k0);
                    v2f wi = *(const v2f*)(wi_p + k0);
                    v2f wf = *(const v2f*)(wf_p + k0);
                    v2f wg = *(const v2f*)(wg_p + k0);
                    v2f wo = *(const v2f*)(wo_p + k0);
                    ai = wmma4(a, wi, ai);
                    af = wmma4(a, wf, af);
                    ag = wmma4(a, wg, ag);
                    ao = wmma4(a, wo, ao);
                }
            }

            // element-wise LSTM cell directly on C-layout accumulators
#pragma unroll
            for (int r = 0; r < 8; ++r) {
                int   m    = r + 8 * lh;                // track row in tile
                int   n    = s * 16 + m_lane;           // hidden column
                float ig   = sigm(ai[r]);
                float fg   = sigm(af[r]);
                float gg   = tanhf(ag[r]);
                float og   = sigm(ao[r]);
                float cold = cs[m][n];
                float cn   = fg * cold + ig * gg;
                float hnew = og * tanhf(cn);
                hn[m][n]   = hnew;                      // unmasked h_new -> ff GEMM
                bool  mk   = (mks[m] != 0);
                cs[m][n]            = mk ? cn   : cold;
                hsbuf[1 - par][m][n] = mk ? hnew : hsbuf[par][m][n];
            }
        }
        par ^= 1;
        __syncthreads();

        // ---- 4. ff = h_new @ ff_w.T + ff_b, NaN-mask rows, stream to HBM ----
#pragma clang loop unroll(disable)
        for (int nt = 0; nt < FFD / 16; ++nt) {
            const int nf = nt * 16 + m_lane;
            v8f acc;
            {
                float bb = ff_b[nf];
#pragma unroll
                for (int r = 0; r < 8; ++r) acc[r] = bb;
            }
            {
                const float* wb_p = ff_w + (size_t)nf * HID + koff;
                const float* ha_p = &hn[m_lane][koff];
#pragma unroll 2
                for (int k0 = 0; k0 < HID; k0 += 4) {
                    v2f a = *(const v2f*)(ha_p + k0);
                    v2f b = *(const v2f*)(wb_p + k0);
                    acc = wmma4(a, b, acc);
                }
            }
#pragma unroll
            for (int r = 0; r < 8; ++r) {
                int m   = r + 8 * lh;
                int trk = track0 + m;
                if (trk < ntracks) {
                    float val = (mks[m] != 0) ? acc[r] : qnan;
                    out[((size_t)t * ntracks + trk) * FFD + nf] = val;
                }
            }
        }
        __syncthreads();   // xs/hn/vel reused next step
    }
}

extern "C" void kernel_launch(void* const* d_in, const int* in_sizes, int n_in,
                              void* d_out, int out_size, void* d_ws, size_t ws_size,
                              hipStream_t stream) {
    const float* observed = (const float*)d_in[0];
    // d_in[1] = goals (unused), d_in[2] = batch_split (unused)
    const float* emb_w = (const float*)d_in[3];
    const float* emb_b = (const float*)d_in[4];
    const float* w_ih  = (const float*)d_in[5];
    const float* w_hh  = (const float*)d_in[6];
    const float* b_ih  = (const float*)d_in[7];
    const float* b_hh  = (const float*)d_in[8];
    const float* ff_w  = (const float*)d_in[9];
    const float* ff_b  = (const float*)d_in[10];

    const int ntracks = in_sizes[1] / 2;                 // goals is (n, 2)
    const int obs_len = in_sizes[0] / (ntracks * 2);     // observed is (T, n, 2)
    const int nsteps  = obs_len - 1;

    const int blocks = (ntracks + 15) / 16;
    lstm_encode_kernel<<<blocks, 32, 0, stream>>>(
        observed, emb_w, emb_b, w_ih, w_hh, b_ih, b_hh, ff_w, ff_b,
        (float*)d_out, ntracks, nsteps);
}